// ModeMultiplexer2D_63660005261658
// MI455X (gfx1250) — compile-verified
//
#include <hip/hip_runtime.h>

typedef __attribute__((ext_vector_type(2))) float v2f;
typedef __attribute__((ext_vector_type(4))) float v4f;
typedef __attribute__((ext_vector_type(8))) float v8f;

// dims
#define BDIM 256
#define XDIM 16
#define YDIM 16
#define FDIM 8
#define PDIM 63

// One workgroup per (u,v,x,y) block: 256 threads = 8 wave32.
// Phase A (waves 0-1 via LDS): H = sum_p theta_p P_p ; U = exp(iH) by
//   A = iH/16, 8th-order Taylor (Horner), 4 squarings.
// Phase B (all waves): W = [[Ur,-Ui],[Ui,Ur]] (16x16 real) applied to
//   1024 stacked [re;im] vectors via v_wmma_f32_16x16x4_f32.
__global__ __launch_bounds__(256)
void ModeMultiplexer2D_63660005261658_kernel(const float* __restrict__ state,
                                             const float* __restrict__ params,
                                             float* __restrict__ out) {
    __shared__ float sTheta[PDIM];
    __shared__ float Ar[72], Ai[72];   // A = iH/16, pitch 9
    __shared__ float Rr[72], Ri[72];   // running product, pitch 9
    __shared__ float Wl[8 * 32];       // W in B-operand register layout

    const int tid = threadIdx.x;
    const int bi  = blockIdx.x;              // ((u*2+v)*16+x)*16+y
    const int y   = bi & 15;
    const int x   = (bi >> 4) & 15;
    const int v   = (bi >> 8) & 1;
    const int u   = (bi >> 9) & 1;

    // ---- stage thetas ----
    if (tid < PDIM) {
        int pbase = (((u * 2 + v) * XDIM + x) * YDIM + y) * PDIM;
        sTheta[tid] = params[pbase + tid];
    }
    __syncthreads();

    const int i = tid >> 3;   // row (only meaningful for tid<64)
    const int j = tid & 7;    // col

    // ---- H accumulation: thread (i,j) owns H[i][j] ----
    if (tid < 64) {
        float hre = 0.f, him = 0.f;
        for (int p = 0; p < PDIM; ++p) {
            int q = p + 1;
            int jt = 0;
            float fre = 1.f, fim = 0.f;
            #pragma unroll
            for (int b = 0; b < 3; ++b) {
                int l  = (q >> (2 * (2 - b))) & 3;
                int ib = (i >> (2 - b)) & 1;
                int cb;
                float gr, gi;
                if (l == 0)      { cb = ib;     gr = 1.f;                gi = 0.f; }
                else if (l == 1) { cb = 1 - ib; gr = 1.f;                gi = 0.f; }
                else if (l == 2) { cb = 1 - ib; gr = 0.f;                gi = ib ? 1.f : -1.f; }
                else             { cb = ib;     gr = ib ? -1.f : 1.f;    gi = 0.f; }
                jt = (jt << 1) | cb;
                float nr = fre * gr - fim * gi;
                float ni = fre * gi + fim * gr;
                fre = nr; fim = ni;
            }
            if (jt == j) {
                float t = sTheta[p];
                hre += t * fre;
                him += t * fim;
            }
        }
        // A = i*H / 16
        Ar[i * 9 + j] = -him * 0.0625f;
        Ai[i * 9 + j] =  hre * 0.0625f;
        Rr[i * 9 + j] = (i == j) ? 1.f : 0.f;
        Ri[i * 9 + j] = 0.f;
    }
    __syncthreads();

    // ---- Horner Taylor: R = I + A*R/k, k = 8..1 ----
    #pragma unroll
    for (int k = 8; k >= 1; --k) {
        float tre = 0.f, tim = 0.f;
        if (tid < 64) {
            #pragma unroll
            for (int kk = 0; kk < 8; ++kk) {
                float ar = Ar[i * 9 + kk], ai = Ai[i * 9 + kk];
                float rr = Rr[kk * 9 + j], ri = Ri[kk * 9 + j];
                tre += ar * rr - ai * ri;
                tim += ar * ri + ai * rr;
            }
        }
        __syncthreads();
        if (tid < 64) {
            float inv = 1.0f / (float)k;
            Rr[i * 9 + j] = ((i == j) ? 1.f : 0.f) + tre * inv;
            Ri[i * 9 + j] = tim * inv;
        }
        __syncthreads();
    }
    // ---- 4 squarings: R = R*R  -> exp(iH) ----
    #pragma unroll
    for (int sq = 0; sq < 4; ++sq) {
        float tre = 0.f, tim = 0.f;
        if (tid < 64) {
            #pragma unroll
            for (int kk = 0; kk < 8; ++kk) {
                float ar = Rr[i * 9 + kk], ai = Ri[i * 9 + kk];
                float rr = Rr[kk * 9 + j], ri = Ri[kk * 9 + j];
                tre += ar * rr - ai * ri;
                tim += ar * ri + ai * rr;
            }
        }
        __syncthreads();
        if (tid < 64) { Rr[i * 9 + j] = tre; Ri[i * 9 + j] = tim; }
        __syncthreads();
    }

    // ---- emit W = [[Ur,-Ui],[Ui,Ur]] in B-operand register layout ----
    // slot jslot, lane L:  value = W[L%16][ 4*(jslot>>1) + (L>>4)*2 + (jslot&1) ]
    {
        int jslot = tid >> 5;
        int L     = tid & 31;
        int f     = L & 15;
        int g     = 4 * (jslot >> 1) + ((L >> 4) << 1) + (jslot & 1);
        int fi = f & 7, gi = g & 7;
        float ur = Rr[fi * 9 + gi], ui = Ri[fi * 9 + gi];
        float val;
        if (f < 8) val = (g < 8) ? ur : -ui;
        else       val = (g < 8) ? ui :  ur;
        Wl[jslot * 32 + L] = val;
    }
    __syncthreads();

    // ---- Phase B: stream 1024 vectors through WMMA ----
    const int lane = tid & 31;
    const int wave = tid >> 5;
    const int hi   = lane >> 4;      // 0: rows {4k,4k+1}, 1: rows {4k+2,4k+3}

    float wv0 = Wl[0 * 32 + lane], wv1 = Wl[1 * 32 + lane];
    float wv2 = Wl[2 * 32 + lane], wv3 = Wl[3 * 32 + lane];
    float wv4 = Wl[4 * 32 + lane], wv5 = Wl[5 * 32 + lane];
    float wv6 = Wl[6 * 32 + lane], wv7 = Wl[7 * 32 + lane];
    v2f b0 = {wv0, wv1}, b1 = {wv2, wv3}, b2 = {wv4, wv5}, b3 = {wv6, wv7};

    const int fp   = lane & 15;
    const int fofs = (fp < 8) ? (2 * fp) : (2 * (fp - 8) + 1);

    for (int tt = 0; tt < 8; ++tt) {
        int tile = wave * 8 + tt;
        // load: lane L & L+16 split vector (tile*16 + L%16)'s 64B chunk
        int vec = tile * 16 + fp;
        int bb = vec >> 2, rr = (vec >> 1) & 1, ss = vec & 1;
        int cbase = ((((bb * XDIM + x) * 4 + (u * 2 + rr)) * YDIM + y) * 4 + (v * 2 + ss)) * 16;
        v4f q0 = *reinterpret_cast<const v4f*>(state + cbase + hi * 4);
        v4f q1 = *reinterpret_cast<const v4f*>(state + cbase + hi * 4 + 8);
        // A chunks (M = vector, K = g'): re0,re1 | re4,re5 | im0,im1 | im4,im5
        v2f a0 = {q0.x, q0.z};
        v2f a1 = {q1.x, q1.z};
        v2f a2 = {q0.y, q0.w};
        v2f a3 = {q1.y, q1.w};

        v8f acc = {};
        acc = __builtin_amdgcn_wmma_f32_16x16x4_f32(false, a0, false, b0, (short)0, acc, false, false);
        acc = __builtin_amdgcn_wmma_f32_16x16x4_f32(false, a1, false, b1, (short)0, acc, false, false);
        acc = __builtin_amdgcn_wmma_f32_16x16x4_f32(false, a2, false, b2, (short)0, acc, false, false);
        acc = __builtin_amdgcn_wmma_f32_16x16x4_f32(false, a3, false, b3, (short)0, acc, false, false);

        // D[M=vec][N=f']: VGPR m -> vector (m + hi*8), column = lane%16 = f'
        #pragma unroll
        for (int m = 0; m < 8; ++m) {
            int vecs = tile * 16 + m + hi * 8;
            int b2i = vecs >> 2, r2 = (vecs >> 1) & 1, s2 = vecs & 1;
            int cb2 = ((((b2i * XDIM + x) * 4 + (u * 2 + r2)) * YDIM + y) * 4 + (v * 2 + s2)) * 16;
            out[cb2 + fofs] = acc[m];
        }
    }
}

extern "C" void kernel_launch(void* const* d_in, const int* in_sizes, int n_in,
                              void* d_out, int out_size, void* d_ws, size_t ws_size,
                              hipStream_t stream) {
    (void)in_sizes; (void)n_in; (void)d_ws; (void)ws_size; (void)out_size;
    const float* state  = (const float*)d_in[0];   // complex64 -> float pairs
    const float* params = (const float*)d_in[1];   // float32 [2,2,16,16,63]
    float* out = (float*)d_out;                    // complex64 -> float pairs
    // 1024 blocks = (u,v,x,y); 256 threads = 8 wave32
    ModeMultiplexer2D_63660005261658_kernel<<<1024, 256, 0, stream>>>(state, params, out);
}